// GraphEmbeddingBackBone_75265006895364
// MI455X (gfx1250) — compile-verified
//
#include <hip/hip_runtime.h>

// GraphEmbeddingBackBone for MI455X (gfx1250).
// Pure data-movement op (ragged gather + pack): ~100 MB traffic, 0 FLOPs ->
// bandwidth-bound (~4.3 us at 23.3 TB/s). WMMA is inapplicable (no
// contraction); the CDNA5 feature that matters is the async LDS<->global
// data-mover path (GLOBAL_LOAD_ASYNC_TO_LDS_B128 /
// GLOBAL_STORE_FROM_LDS_B128 tracked by ASYNCcnt), which streams rows without
// a VGPR round trip. This revision pipelines 4 rows per block so each wave
// keeps 4 x 128B async transactions in flight before its single
// s_wait_asynccnt, and relies on S_ENDPGM's implicit wait-idle to cover the
// final stores.

#define MAX_NODE 511
#define SLOTS (MAX_NODE + 1)
#define ROWS_PER_BLOCK 4
#define ROW_GROUPS ((MAX_NODE + ROWS_PER_BLOCK - 1) / ROWS_PER_BLOCK)  // 128

// ---------------------------------------------------------------------------
// Kernel 0: exclusive cumsum of num_nodes[B] -> offsets[B] in workspace.
// B == 32, trivial serial scan.
// ---------------------------------------------------------------------------
__global__ void offsets_kernel(const int* __restrict__ num_nodes,
                               int* __restrict__ offsets, int B) {
    if (blockIdx.x == 0 && threadIdx.x == 0) {
        int acc = 0;
        for (int b = 0; b < B; ++b) {
            offsets[b] = acc;
            acc += num_nodes[b];
        }
    }
}

// ---------------------------------------------------------------------------
// Kernel 1: mask, pos_x, pos_y, and features[b, 0, :] = image_embedding[b, :].
// One block per graph, SLOTS (=512) threads: one output slot per thread.
// ---------------------------------------------------------------------------
__global__ __launch_bounds__(SLOTS) void small_kernel(
    const float* __restrict__ centroids,
    const float* __restrict__ image_embedding,
    const int* __restrict__ num_nodes,
    const int* __restrict__ offsets,
    float* __restrict__ features,
    float* __restrict__ mask,
    float* __restrict__ pos_x,
    float* __restrict__ pos_y,
    int C) {
    const int b = blockIdx.x;
    const int s = threadIdx.x;  // slot index, 0..SLOTS-1
    const int n = num_nodes[b];
    const int off = offsets[b];

    // mask[b, s] = (s >= n+1) && (n <= MAX_NODE)
    mask[(size_t)b * SLOTS + s] =
        ((s >= n + 1) && (n <= MAX_NODE)) ? 1.0f : 0.0f;

    // pos_x / pos_y: slot 0 is the image slot (0.0); others gather centroids
    float px = 0.0f, py = 0.0f;
    if (s > 0) {
        const int j = s - 1;
        const int n_eff = (n < MAX_NODE) ? n : MAX_NODE;
        if (j < n_eff) {
            const int idx = off + j;
            px = centroids[2 * (size_t)idx + 0];
            py = centroids[2 * (size_t)idx + 1];
        }
    }
    pos_x[(size_t)b * SLOTS + s] = px;
    pos_y[(size_t)b * SLOTS + s] = py;

    // features[b, 0, :] = image_embedding[b, :]   (C floats, float2 per lane)
    const float2* src = (const float2*)(image_embedding + (size_t)b * C);
    float2*       dst = (float2*)(features + (size_t)b * SLOTS * C);
    for (int i = s; i < C / 2; i += SLOTS) dst[i] = src[i];
}

// ---------------------------------------------------------------------------
// Kernel 2: the big gather. One 256-thread block per group of 4 node rows of
// one graph. For each valid row we stage its 4 KB through LDS with CDNA5
// async ops; invalid rows are zero-filled with direct b128 stores while the
// async loads are in flight. All row predicates are block-uniform, so EXEC
// stays all-ones on every async op, and each wave's lanes only touch their
// own LDS bytes (no barrier needed — only the per-wave s_wait_asynccnt).
// Pipeline: [issue 4 async loads] -> [zero-fill invalid rows] ->
//           [s_wait_asynccnt 0] -> [issue 4 async stores] -> S_ENDPGM
//           (implicit wait-idle covers store completion).
// ---------------------------------------------------------------------------
__global__ __launch_bounds__(256) void gather_kernel(
    const float* __restrict__ patch,
    const int* __restrict__ num_nodes,
    const int* __restrict__ offsets,
    float* __restrict__ features,
    int C) {
    __shared__ float4 buf[ROWS_PER_BLOCK][256];  // 16 KB staging

    const int b = blockIdx.x / ROW_GROUPS;
    const int j0 = (blockIdx.x % ROW_GROUPS) * ROWS_PER_BLOCK;
    const int tid = threadIdx.x;

    const int n = num_nodes[b];
    const int n_eff = (n < MAX_NODE) ? n : MAX_NODE;
    const int off = offsets[b];

    // One 16-byte chunk per lane per row (C == 1024 -> 256 chunks per row).
    if (tid * 4 >= C) return;

    bool any_valid = false;

    // Phase 1: issue async global->LDS loads for all valid rows (up to 4 in
    // flight per wave before any wait).
    #pragma unroll
    for (int r = 0; r < ROWS_PER_BLOCK; ++r) {
        const int j = j0 + r;
        if (j < MAX_NODE && j < n_eff) {
            any_valid = true;
            const float* src = patch + (size_t)(off + j) * C + 4 * (size_t)tid;
            const unsigned lds_addr = (unsigned)(size_t)(&buf[r][tid]);
            const unsigned long long sa = (unsigned long long)(size_t)src;
            asm volatile("global_load_async_to_lds_b128 %0, %1, off"
                         :
                         : "v"(lds_addr), "v"(sa)
                         : "memory");
        }
    }

    // Phase 2: zero-fill invalid rows directly while the loads are in flight.
    const float4 z = make_float4(0.0f, 0.0f, 0.0f, 0.0f);
    #pragma unroll
    for (int r = 0; r < ROWS_PER_BLOCK; ++r) {
        const int j = j0 + r;
        if (j < MAX_NODE && j >= n_eff) {
            float* dst = features + ((size_t)b * SLOTS + 1 + (size_t)j) * C +
                         4 * (size_t)tid;
            *(float4*)dst = z;
        }
    }

    // Phase 3: wait for all loads (in-order completion), then issue the
    // async LDS->global stores. No trailing wait: S_ENDPGM waits idle.
    if (any_valid) {
        asm volatile("s_wait_asynccnt 0" ::: "memory");
        #pragma unroll
        for (int r = 0; r < ROWS_PER_BLOCK; ++r) {
            const int j = j0 + r;
            if (j < MAX_NODE && j < n_eff) {
                float* dst = features +
                             ((size_t)b * SLOTS + 1 + (size_t)j) * C +
                             4 * (size_t)tid;
                const unsigned lds_addr = (unsigned)(size_t)(&buf[r][tid]);
                const unsigned long long da = (unsigned long long)(size_t)dst;
                asm volatile("global_store_async_from_lds_b128 %0, %1, off"
                             :
                             : "v"(da), "v"(lds_addr)
                             : "memory");
            }
        }
    }
}

// ---------------------------------------------------------------------------
// Launch wrapper.
// d_in order: patch_embeddings [T*C] f32, centroids [T*2] f32,
//             image_embedding [B*C] f32, num_nodes [B] i32.
// d_out: features [B*SLOTS*C] ++ mask [B*SLOTS] ++ pos_x ++ pos_y (f32).
// ---------------------------------------------------------------------------
extern "C" void kernel_launch(void* const* d_in, const int* in_sizes, int n_in,
                              void* d_out, int out_size, void* d_ws,
                              size_t ws_size, hipStream_t stream) {
    const float* patch     = (const float*)d_in[0];
    const float* centroids = (const float*)d_in[1];
    const float* image     = (const float*)d_in[2];
    const int*   num_nodes = (const int*)d_in[3];

    const int B = in_sizes[3];          // 32
    const int C = in_sizes[2] / B;      // 1024

    float* features = (float*)d_out;
    float* mask  = features + (size_t)B * SLOTS * C;
    float* pos_x = mask + (size_t)B * SLOTS;
    float* pos_y = pos_x + (size_t)B * SLOTS;

    int* offsets = (int*)d_ws;

    offsets_kernel<<<1, 32, 0, stream>>>(num_nodes, offsets, B);
    small_kernel<<<B, SLOTS, 0, stream>>>(centroids, image, num_nodes, offsets,
                                          features, mask, pos_x, pos_y, C);
    gather_kernel<<<B * ROW_GROUPS, 256, 0, stream>>>(patch, num_nodes,
                                                      offsets, features, C);
}